// Line_block_41188736369115
// MI455X (gfx1250) — compile-verified
//
#include <hip/hip_runtime.h>
#include <math.h>

typedef float v2f __attribute__((ext_vector_type(2)));
typedef float v8f __attribute__((ext_vector_type(8)));

#define BB 8
#define GG 2048
#define KK 32
#define CC 64
#define TWO_PI 6.28318530717958647692f

__device__ __forceinline__ float wave_allreduce_sum(float v) {
    v += __shfl_xor(v, 1, 32);
    v += __shfl_xor(v, 2, 32);
    v += __shfl_xor(v, 4, 32);
    v += __shfl_xor(v, 8, 32);
    v += __shfl_xor(v, 16, 32);
    return v;
}

// One wave32 per (b,g) point.  Computes:
//   invn[b*G+g]          = 1/||lc_x[b,g,:]||          (C=64 reduction)
//   klbuf[(b*G+g)*64+c]  = 2*mean_k( f(x_proj)[k,c] ) (the un-normalized kl)
// x_proj = knn_line(32x7) @ fourier_B(7x32) done with V_WMMA_F32_16X16X4_F32,
// K padded 7->8, tiled 2x2x2 (M,N,K) = 8 WMMA ops per wave.
__global__ __launch_bounds__(256)
void k1_fourier_wmma(const float* __restrict__ lc_x,
                     const float* __restrict__ knn_x,
                     const float* __restrict__ fB,
                     float* __restrict__ klbuf,
                     float* __restrict__ invn)
{
    const int tid  = threadIdx.x;
    const int lane = tid & 31;
    const int wid  = (blockIdx.x * blockDim.x + tid) >> 5;   // 0 .. B*G-1
    const int hi   = lane >> 4;                              // lane half
    const int c15  = lane & 15;

    const float* lcp = lc_x + (size_t)wid * CC;

    // ---- 1/||lc_x|| over C=64 (wave butterfly reduce) ----
    float a0 = lcp[lane];
    float a1 = lcp[lane + 32];
    float ss = wave_allreduce_sum(a0 * a0 + a1 * a1);
    float inv = rsqrtf(ss);
    if (lane == 0) invn[wid] = inv;

    // ---- local normal n = lc_x[b,g,3:6] ----
    const float nx = lcp[3], ny = lcp[4], nz = lcp[5];

    // ---- A matrix (knn_line, 32x8): ISA 16x4 f32 layout.
    // Tile (m,t): lane holds row M = c15 + 16m; VGPR pair holds
    // K = 4t + 2*hi and 4t + 2*hi + 1.  Components:
    //   K: 0=px 1=py 2=pz 3=cx 4=cy 5=cz 6=dot 7=pad0
    const float* knp = knn_x + (size_t)wid * (KK * 6);
    v2f A[2][2];
#pragma unroll
    for (int m = 0; m < 2; ++m) {
        const int k = c15 + 16 * m;
        const float* kp = knp + k * 6;
        const float px = kp[3], py = kp[4], pz = kp[5];
        const float cx = py * nz - pz * ny;
        const float cy = pz * nx - px * nz;
        const float cz = px * ny - py * nx;
        const float dt = px * nx + py * ny + pz * nz;
        A[m][0].x = hi ? pz : px;
        A[m][0].y = hi ? cx : py;
        A[m][1].x = hi ? dt : cy;
        A[m][1].y = hi ? 0.f : cz;
    }

    // ---- B matrix (fourier_B, 8x32, row 7 = 0): 4x16 f32 layout,
    // lane holds column N = 16n + c15; K split across lane halves.
    v2f Bm[2][2];
#pragma unroll
    for (int n = 0; n < 2; ++n) {
        const int N = 16 * n + c15;
        Bm[n][0].x = hi ? fB[2 * KK + N] : fB[0 * KK + N];
        Bm[n][0].y = hi ? fB[3 * KK + N] : fB[1 * KK + N];
        Bm[n][1].x = hi ? fB[6 * KK + N] : fB[4 * KK + N];
        Bm[n][1].y = hi ? 0.f            : fB[5 * KK + N];
    }

    // ---- GEMM + fused sin^5/cos^5 + K-reduction ----
    float ssum[2] = {0.f, 0.f};
    float csum[2] = {0.f, 0.f};
#pragma unroll
    for (int n = 0; n < 2; ++n) {
#pragma unroll
        for (int m = 0; m < 2; ++m) {
            v8f d = {};
            d = __builtin_amdgcn_wmma_f32_16x16x4_f32(
                    false, A[m][0], false, Bm[n][0], (short)0, d, false, false);
            d = __builtin_amdgcn_wmma_f32_16x16x4_f32(
                    false, A[m][1], false, Bm[n][1], (short)0, d, false, false);
#pragma unroll
            for (int v = 0; v < 8; ++v) {
                const float x = TWO_PI * d[v];
                const float s = __sinf(x);
                const float c = __cosf(x);
                const float s2 = s * s, c2 = c * c;
                ssum[n] += s2 * s2 * s;   // s^5 (odd power keeps sign)
                csum[n] += c2 * c2 * c;
            }
        }
    }
    // merge the two M lane-halves -> full sum over all 32 k
#pragma unroll
    for (int n = 0; n < 2; ++n) {
        ssum[n] += __shfl_xor(ssum[n], 16, 32);
        csum[n] += __shfl_xor(csum[n], 16, 32);
    }

    // kl_raw = 2 * mean_k = sum / 16
    float* outp = klbuf + (size_t)wid * CC;
    if (hi == 0) {
        outp[c15]      = ssum[0] * 0.0625f;   // sin, cols 0..15
        outp[16 + c15] = ssum[1] * 0.0625f;   // sin, cols 16..31
    } else {
        outp[32 + c15] = csum[0] * 0.0625f;   // cos, cols 0..15
        outp[48 + c15] = csum[1] * 0.0625f;   // cos, cols 16..31
    }
}

// One block per (b,c): L2-normalize kl over g, emit
//   out[b, c,     g] = kl_hat - lc_line
//   out[b, 64+c,  g] = lc_line
__global__ __launch_bounds__(256)
void k2_normalize(const float* __restrict__ lc_x,
                  const float* __restrict__ klbuf,
                  const float* __restrict__ invn,
                  float* __restrict__ out)
{
    const int bc  = blockIdx.x;        // 0 .. B*C-1
    const int b   = bc >> 6;
    const int c   = bc & 63;
    const int tid = threadIdx.x;

    __shared__ float sred[256];
    __shared__ float s_inv;

    float acc = 0.f;
    for (int g = tid; g < GG; g += 256) {
        const float v = klbuf[((size_t)(b * GG + g)) * CC + c];
        acc += v * v;
    }
    sred[tid] = acc;
    __syncthreads();
    for (int s = 128; s > 0; s >>= 1) {
        if (tid < s) sred[tid] += sred[tid + s];
        __syncthreads();
    }
    if (tid == 0) s_inv = rsqrtf(sred[0]);
    __syncthreads();
    const float ikl = s_inv;

    for (int g = tid; g < GG; g += 256) {
        const size_t idx = ((size_t)(b * GG + g)) * CC + c;
        const float kn = klbuf[idx] * ikl;
        const float ll = lc_x[idx] * invn[b * GG + g];
        out[((size_t)(b * 128 + c)) * GG + g]      = kn - ll;
        out[((size_t)(b * 128 + 64 + c)) * GG + g] = ll;
    }
}

extern "C" void kernel_launch(void* const* d_in, const int* in_sizes, int n_in,
                              void* d_out, int out_size, void* d_ws, size_t ws_size,
                              hipStream_t stream) {
    const float* lc_x  = (const float*)d_in[0];   // (8,2048,64)
    const float* knn_x = (const float*)d_in[1];   // (8,2048,32,6)
    const float* fB    = (const float*)d_in[2];   // (7,32)
    float* out = (float*)d_out;                   // (8,128,2048)

    float* klbuf = (float*)d_ws;                       // B*G*64 floats = 4 MB
    float* invn  = klbuf + (size_t)BB * GG * CC;       // B*G floats

    // 16384 waves, 8 waves (256 threads) per block
    k1_fourier_wmma<<<(BB * GG) / 8, 256, 0, stream>>>(lc_x, knn_x, fB, klbuf, invn);
    k2_normalize<<<BB * CC, 256, 0, stream>>>(lc_x, klbuf, invn, out);
}